// Net_44547400794679
// MI455X (gfx1250) — compile-verified
//
#include <hip/hip_runtime.h>
#include <math.h>

// ---- problem constants (fixed by the reference) ----
#define N0c 131072
#define N1c 65536
#define N2c 32768
#define N3c 16384
#define B8c 16384
#define E1c 1703936
#define E2c 851968
#define E3c 425984
#define NBc 2048   // batch rows for FC (B8*32 / 256)

typedef float v2f __attribute__((ext_vector_type(2)));
typedef float v8f __attribute__((ext_vector_type(8)));

// ---------- helpers ----------
__device__ __forceinline__ unsigned fenc(float f) {
  unsigned u = __float_as_uint(f);
  return (u & 0x80000000u) ? ~u : (u | 0x80000000u);   // order-preserving map
}
__device__ __forceinline__ float fdec(unsigned e) {
  unsigned u = (e & 0x80000000u) ? (e & 0x7fffffffu) : ~e;
  return __uint_as_float(u);
}
__device__ __forceinline__ float elu1(float v) { return v > 0.f ? v : expf(v) - 1.f; }

struct Basis { float w[8]; int flat[8]; };

// degree-1 open B-spline basis, K=5 -> v = p*4, 8 (weight, bin) corners
__device__ __forceinline__ Basis spline_basis(const float* __restrict__ ps, int e) {
  float v0 = ps[e * 3 + 0] * 4.f, v1 = ps[e * 3 + 1] * 4.f, v2 = ps[e * 3 + 2] * 4.f;
  float k0 = fminf(fmaxf(floorf(v0), 0.f), 3.f);
  float k1 = fminf(fmaxf(floorf(v1), 0.f), 3.f);
  float k2 = fminf(fmaxf(floorf(v2), 0.f), 3.f);
  float f0 = v0 - k0, f1 = v1 - k1, f2 = v2 - k2;
  int i0 = (int)k0, i1 = (int)k1, i2 = (int)k2;
  Basis b;
#pragma unroll
  for (int c = 0; c < 8; ++c) {
    int o0 = c & 1, o1 = (c >> 1) & 1, o2 = (c >> 2) & 1;
    b.w[c] = (o0 ? f0 : 1.f - f0) * (o1 ? f1 : 1.f - f1) * (o2 ? f2 : 1.f - f2);
    b.flat[c] = ((i0 + o0) * 5 + (i1 + o1)) * 5 + (i2 + o2);
  }
  return b;
}

// ---------- segment-max pooling ----------
__global__ void pool_scatter(const float* __restrict__ in, const int* __restrict__ cluster,
                             unsigned* __restrict__ enc, int nIn, int C) {
  int tid = blockIdx.x * blockDim.x + threadIdx.x;
  if (tid >= nIn * C) return;
  int node = tid / C, ch = tid - node * C;
  atomicMax(&enc[cluster[node] * C + ch], fenc(in[tid]));
}

// decode in place; untouched slots (enc==0) decode to NaN -> 0 (matches reference)
__global__ void pool_decode(unsigned* __restrict__ buf, int n) {
  int tid = blockIdx.x * blockDim.x + threadIdx.x;
  if (tid >= n) return;
  float v = fdec(buf[tid]);
  unsigned u = __float_as_uint(v);
  if ((u & 0x7f800000u) == 0x7f800000u) v = 0.f;  // non-finite -> 0
  ((float*)buf)[tid] = v;
}

// ---------- spline conv, cin = 1 (layer 1): wave per 8 edges, lane = cout ----------
__global__ void conv_edges1(const float* __restrict__ x, const int* __restrict__ ei,
                            const float* __restrict__ ps, const float* __restrict__ W1,
                            float* __restrict__ agg, float* __restrict__ deg, int E) {
  int gid = blockIdx.x * blockDim.x + threadIdx.x;
  int lane = gid & 31;
  int e = (gid >> 5) * 8;
  int eEnd = e + 8 < E ? e + 8 : E;
  for (; e < eEnd; ++e) {
    int src = ei[e], dst = ei[E + e];
    Basis b = spline_basis(ps, e);
    float xj = x[src];
    float s = 0.f;
#pragma unroll
    for (int c = 0; c < 8; ++c)
      s = fmaf(b.w[c], W1[b.flat[c] * 32 + lane], s);
    atomicAdd(&agg[dst * 32 + lane], xj * s);
    if (lane == 0) atomicAdd(&deg[dst], 1.f);
  }
}

// ---------- spline conv, cin = cout = 32: one wave per edge, lane = cout ----------
__global__ void conv_edges32(const float* __restrict__ x, const int* __restrict__ ei,
                             const float* __restrict__ ps, const float* __restrict__ W,
                             float* __restrict__ agg, float* __restrict__ deg, int E) {
  int gid = blockIdx.x * blockDim.x + threadIdx.x;
  int e = gid >> 5, lane = gid & 31;
  if (e >= E) return;
  int src = ei[e], dst = ei[E + e];
  Basis b = spline_basis(ps, e);
  float xj = x[src * 32 + lane];           // lane holds x_j[lane]; broadcast via shuffle
  float m = 0.f;
#pragma unroll
  for (int c = 0; c < 8; ++c) {
    const float* Wc = W + b.flat[c] * 1024 + lane;  // W[flat][i][o=lane], coalesced per i
    float acc = 0.f;
#pragma unroll
    for (int i = 0; i < 32; ++i)
      acc = fmaf(__shfl(xj, i, 32), Wc[i * 32], acc);
    m = fmaf(b.w[c], acc, m);
  }
  atomicAdd(&agg[dst * 32 + lane], m);
  if (lane == 0) atomicAdd(&deg[dst], 1.f);
}

// ---------- finalize layer 1 (root is rank-1, elementwise) ----------
__global__ void finalize1(const float* __restrict__ h, const float* __restrict__ agg,
                          const float* __restrict__ deg, const float* __restrict__ root1,
                          const float* __restrict__ b1, float* __restrict__ out, int n) {
  int tid = blockIdx.x * blockDim.x + threadIdx.x;
  if (tid >= n * 32) return;
  int nd = tid >> 5, o = tid & 31;
  float v = agg[tid] / fmaxf(deg[nd], 1.f) + h[nd] * root1[o] + b1[o];
  out[tid] = elu1(v);
}

// ---------- finalize 32-ch layers: root GEMM via V_WMMA_F32_16X16X4_F32 ----------
// One wave handles 16 nodes: C(16x32) = X(16x32) @ root(32x32), K stepped by 4,
// two 16-wide cout tiles as two accumulators. Then agg/deg/bias + ELU.
__global__ void finalize_conv32(const float* __restrict__ xin, const float* __restrict__ agg,
                                const float* __restrict__ deg, const float* __restrict__ root,
                                const float* __restrict__ bias, float* __restrict__ out, int n) {
  int gid = blockIdx.x * blockDim.x + threadIdx.x;
  int wave = gid >> 5, lane = gid & 31;
  int row0 = wave * 16;
  if (row0 >= n) return;                 // wave-uniform: EXEC stays all-ones for WMMA
  int m = lane & 15, half = lane >> 4;
  v8f c0 = {}; v8f c1 = {};
#pragma unroll
  for (int k0 = 0; k0 < 32; k0 += 4) {
    int kr = k0 + 2 * half;              // A/B half-wave K split per ISA 16x4 layout
    v2f a; a.x = xin[(row0 + m) * 32 + kr];
           a.y = xin[(row0 + m) * 32 + kr + 1];
    v2f b0; b0.x = root[kr * 32 + m];        b0.y = root[(kr + 1) * 32 + m];
    v2f b1; b1.x = root[kr * 32 + 16 + m];   b1.y = root[(kr + 1) * 32 + 16 + m];
    c0 = __builtin_amdgcn_wmma_f32_16x16x4_f32(false, a, false, b0, (short)0, c0, false, false);
    c1 = __builtin_amdgcn_wmma_f32_16x16x4_f32(false, a, false, b1, (short)0, c1, false, false);
  }
  // C layout: VGPR r -> rows (r, r+8) across half-waves, N = lane&15
#pragma unroll
  for (int r = 0; r < 8; ++r) {
    int row = row0 + r + half * 8;
    float d = fmaxf(deg[row], 1.f);
    float v0 = agg[row * 32 + m]      / d + c0[r] + bias[m];
    float v1 = agg[row * 32 + 16 + m] / d + c1[r] + bias[16 + m];
    out[row * 32 + m]      = elu1(v0);
    out[row * 32 + 16 + m] = elu1(v1);
  }
}

// ---------- FC: [2048,256] @ [256,10] via WMMA, logits to scratch [2048,16] ----------
__global__ void fc_wmma(const float* __restrict__ h, const float* __restrict__ fcw,
                        const float* __restrict__ fcb, float* __restrict__ logits) {
  int gid = blockIdx.x * blockDim.x + threadIdx.x;
  int wave = gid >> 5, lane = gid & 31;
  int row0 = wave * 16;
  if (row0 >= NBc) return;
  int m = lane & 15, half = lane >> 4;
  v8f c = {};
  for (int k0 = 0; k0 < 256; k0 += 4) {
    int kr = k0 + 2 * half;
    v2f a; a.x = h[(row0 + m) * 256 + kr];
           a.y = h[(row0 + m) * 256 + kr + 1];
    v2f b; b.x = (m < 10) ? fcw[kr * 10 + m] : 0.f;
           b.y = (m < 10) ? fcw[(kr + 1) * 10 + m] : 0.f;
    c = __builtin_amdgcn_wmma_f32_16x16x4_f32(false, a, false, b, (short)0, c, false, false);
  }
#pragma unroll
  for (int r = 0; r < 8; ++r) {
    int row = row0 + r + half * 8;
    logits[row * 16 + m] = c[r] + ((m < 10) ? fcb[m] : 0.f);
  }
}

__global__ void log_softmax10(const float* __restrict__ logits, float* __restrict__ out) {
  int row = blockIdx.x * blockDim.x + threadIdx.x;
  if (row >= NBc) return;
  float v[10], mx = -1e30f;
#pragma unroll
  for (int j = 0; j < 10; ++j) { v[j] = logits[row * 16 + j]; mx = fmaxf(mx, v[j]); }
  float s = 0.f;
#pragma unroll
  for (int j = 0; j < 10; ++j) s += expf(v[j] - mx);
  float lse = mx + logf(s);
#pragma unroll
  for (int j = 0; j < 10; ++j) out[row * 10 + j] = v[j] - lse;
}

// ---------------------------------------------------------------------------
extern "C" void kernel_launch(void* const* d_in, const int* in_sizes, int n_in,
                              void* d_out, int out_size, void* d_ws, size_t ws_size,
                              hipStream_t stream) {
  const float* x        = (const float*)d_in[0];
  const int*   cluster0 = (const int*)  d_in[1];
  const int*   ei1      = (const int*)  d_in[2];
  const float* ps1      = (const float*)d_in[3];
  const int*   cluster1 = (const int*)  d_in[4];
  const int*   ei2      = (const int*)  d_in[5];
  const float* ps2      = (const float*)d_in[6];
  const int*   cluster2 = (const int*)  d_in[7];
  const int*   ei3      = (const int*)  d_in[8];
  const float* ps3      = (const float*)d_in[9];
  const int*   cluster3 = (const int*)  d_in[10];
  const float* W1 = (const float*)d_in[11];
  const float* r1 = (const float*)d_in[12];
  const float* b1 = (const float*)d_in[13];
  const float* W2 = (const float*)d_in[14];
  const float* r2 = (const float*)d_in[15];
  const float* b2 = (const float*)d_in[16];
  const float* W3 = (const float*)d_in[17];
  const float* r3 = (const float*)d_in[18];
  const float* b3 = (const float*)d_in[19];
  const float* fcw = (const float*)d_in[20];
  const float* fcb = (const float*)d_in[21];
  float* out = (float*)d_out;

  // ---- workspace layout (floats); total ~9.7M elems (~39 MB), fits L2 entirely ----
  float* ws = (float*)d_ws;
  size_t off = 0;
  float* h1   = ws + off; off += N1c;          // pool0 enc/decoded, [N1,1]
  float* agg1 = ws + off; off += (size_t)N1c * 32;
  float* deg1 = ws + off; off += N1c;
  float* out1 = ws + off; off += (size_t)N1c * 32;
  float* h2   = ws + off; off += (size_t)N2c * 32;
  float* agg2 = ws + off; off += (size_t)N2c * 32;
  float* deg2 = ws + off; off += N2c;
  float* out2 = ws + off; off += (size_t)N2c * 32;
  float* h3   = ws + off; off += (size_t)N3c * 32;
  float* agg3 = ws + off; off += (size_t)N3c * 32;
  float* deg3 = ws + off; off += N3c;
  float* out3 = ws + off; off += (size_t)N3c * 32;
  float* h4   = ws + off; off += (size_t)B8c * 32;
  float* lgt  = ws + off; off += (size_t)NBc * 16;
  hipMemsetAsync(d_ws, 0, off * sizeof(float), stream);  // zero aggs/degs + pool enc init

  const int T = 256;
  auto blk = [](long n) { return (unsigned)((n + 255) / 256); };

  // level 1: pool(x -> N1), conv1, finalize
  pool_scatter<<<blk((long)N0c), T, 0, stream>>>(x, cluster0, (unsigned*)h1, N0c, 1);
  pool_decode <<<blk((long)N1c), T, 0, stream>>>((unsigned*)h1, N1c);
  conv_edges1 <<<blk(((long)(E1c + 7) / 8) * 32), T, 0, stream>>>(h1, ei1, ps1, W1, agg1, deg1, E1c);
  finalize1   <<<blk((long)N1c * 32), T, 0, stream>>>(h1, agg1, deg1, r1, b1, out1, N1c);

  // level 2
  pool_scatter<<<blk((long)N1c * 32), T, 0, stream>>>(out1, cluster1, (unsigned*)h2, N1c, 32);
  pool_decode <<<blk((long)N2c * 32), T, 0, stream>>>((unsigned*)h2, N2c * 32);
  conv_edges32<<<blk((long)E2c * 32), T, 0, stream>>>(h2, ei2, ps2, W2, agg2, deg2, E2c);
  finalize_conv32<<<blk(((long)N2c / 16) * 32), T, 0, stream>>>(h2, agg2, deg2, r2, b2, out2, N2c);

  // level 3
  pool_scatter<<<blk((long)N2c * 32), T, 0, stream>>>(out2, cluster2, (unsigned*)h3, N2c, 32);
  pool_decode <<<blk((long)N3c * 32), T, 0, stream>>>((unsigned*)h3, N3c * 32);
  conv_edges32<<<blk((long)E3c * 32), T, 0, stream>>>(h3, ei3, ps3, W3, agg3, deg3, E3c);
  finalize_conv32<<<blk(((long)N3c / 16) * 32), T, 0, stream>>>(h3, agg3, deg3, r3, b3, out3, N3c);

  // final pool -> [2048,256], FC via WMMA, log_softmax
  pool_scatter<<<blk((long)N3c * 32), T, 0, stream>>>(out3, cluster3, (unsigned*)h4, N3c, 32);
  pool_decode <<<blk((long)B8c * 32), T, 0, stream>>>((unsigned*)h4, B8c * 32);
  fc_wmma     <<<blk(((long)NBc / 16) * 32), T, 0, stream>>>(h4, fcw, fcb, lgt);
  log_softmax10<<<blk((long)NBc), T, 0, stream>>>(lgt, out);
}